// PositionalEncoding3D_22247930593290
// MI455X (gfx1250) — compile-verified
//
#include <hip/hip_runtime.h>
#include <hip/hip_bf16.h>

typedef __attribute__((ext_vector_type(2))) float v2f;
typedef __attribute__((ext_vector_type(4))) float v4f;
typedef __attribute__((ext_vector_type(8))) float v8f;

#define CDIM 64
#define ROWS_PER_WAVE 16
#define NWAVES 8
#define ROWS_PER_BLOCK (ROWS_PER_WAVE * NWAVES)   // 128
#define BLOCK_THREADS 256                          // 8 wave32

// out[b,n,c] = features[b,n,c] + (pe_flat @ W^T)[n,c] + bias[c]
// One pass: each block computes proj for its 128 rows via V_WMMA_F32_16X16X4_F32
// into LDS, then streams all batches with 16-byte NT loads/stores.
__global__ __launch_bounds__(BLOCK_THREADS)
void PositionalEncoding3D_kernel(const float* __restrict__ features,
                                 const float* __restrict__ pe,
                                 const float* __restrict__ W,
                                 const float* __restrict__ bias,
                                 float* __restrict__ out,
                                 int N, int Bn) {
    __shared__ __align__(16) float lds_proj[ROWS_PER_BLOCK * CDIM];   // 32 KB

    const int tid    = threadIdx.x;
    const int wave   = tid >> 5;        // wave32
    const int lane   = tid & 31;
    const int lane16 = lane & 15;
    const int hi     = lane >> 4;       // 0: lanes 0-15, 1: lanes 16-31
    const int koff   = hi * 2;          // K sub-offset within a k-step of 4

    const int block_row0 = blockIdx.x * ROWS_PER_BLOCK;
    const int wave_row0  = block_row0 + wave * ROWS_PER_WAVE;

    // ---- Phase 1: proj tile = A(16x64) @ W^T(64x64), via 16x16x4 f32 WMMA ----
    // A layout (32-bit A 16x4 per step): lane m<16 -> row m, K = 4s+{0,1};
    // lanes 16-31 -> row m, K = 4s+{2,3}.
    const float* arow = pe + (size_t)(wave_row0 + lane16) * CDIM;
    const v2f*   a2   = (const v2f*)(arow + koff);        // 8B aligned (row stride 256B)
    v2f A[16];
#pragma unroll
    for (int s = 0; s < 16; ++s) {
        A[s] = a2[2 * s];               // floats at 4s+koff, 4s+koff+1
    }

#pragma unroll
    for (int jt = 0; jt < 4; ++jt) {
        // B layout (4x16 per step): lane holds column N = jt*16 + lane16;
        // B[k][n] = W^T[k][n] = W[n*64 + k]  -> same striding pattern as A.
        const float* wcol = W + (size_t)(jt * 16 + lane16) * CDIM;
        const v2f*   w2   = (const v2f*)(wcol + koff);
        v8f acc = {};
#pragma unroll
        for (int s = 0; s < 16; ++s) {
            v2f Bv = w2[2 * s];
            acc = __builtin_amdgcn_wmma_f32_16x16x4_f32(
                /*neg_a=*/false, A[s], /*neg_b=*/false, Bv,
                /*c_mod=*/(short)0, acc, /*reuse_a=*/false, /*reuse_b=*/false);
        }
        // D layout: VGPR i -> row (i + hi*8), col = jt*16 + lane16. Fold bias here.
        const int   col = jt * 16 + lane16;
        const float bi  = bias[col];
#pragma unroll
        for (int i = 0; i < 8; ++i) {
            const int lrow = wave * ROWS_PER_WAVE + i + hi * 8;
            lds_proj[lrow * CDIM + col] = acc[i] + bi;
        }
    }
    __syncthreads();

    // ---- Phase 2: stream every batch: out = features + proj (coalesced 16B) ----
    const size_t tile_base = (size_t)block_row0 * CDIM;
    const int    n_vec4    = (ROWS_PER_BLOCK * CDIM) / 4;      // 2048
    for (int b = 0; b < Bn; ++b) {
        const v4f* fsrc = (const v4f*)(features + (size_t)b * N * CDIM + tile_base);
        v4f*       dst  = (v4f*)(out + (size_t)b * N * CDIM + tile_base);
        const v4f* psrc = (const v4f*)lds_proj;
#pragma unroll
        for (int v = 0; v < n_vec4 / BLOCK_THREADS; ++v) {     // 8 iterations
            const int idx = v * BLOCK_THREADS + tid;
            v4f f = __builtin_nontemporal_load(&fsrc[idx]);
            v4f p = psrc[idx];
            v4f o = f + p;
            __builtin_nontemporal_store(o, &dst[idx]);
        }
    }
}

extern "C" void kernel_launch(void* const* d_in, const int* in_sizes, int n_in,
                              void* d_out, int out_size, void* d_ws, size_t ws_size,
                              hipStream_t stream) {
    const float* features = (const float*)d_in[0];   // [B, N, 64] f32
    const float* pe       = (const float*)d_in[1];   // [X, Y, Z, 64] == [N, 64] f32
    const float* W        = (const float*)d_in[2];   // [64, 64] f32
    const float* bias     = (const float*)d_in[3];   // [64] f32
    float* out = (float*)d_out;

    const int N  = in_sizes[1] / CDIM;               // 131072
    const int Bn = in_sizes[0] / in_sizes[1];        // 8

    const int grid = N / ROWS_PER_BLOCK;             // 1024 blocks
    PositionalEncoding3D_kernel<<<grid, BLOCK_THREADS, 0, stream>>>(
        features, pe, W, bias, out, N, Bn);
}